// NCEAverage_87729001988855
// MI455X (gfx1250) — compile-verified
//
#include <hip/hip_runtime.h>
#include <math.h>

// Problem constants (match reference)
#define BATCH 128
#define DIM   128
#define NMEM  200000
#define K1    2048
#define TNORM 0.07f
#define OUTS_ELEMS (6LL * BATCH * K1)      // 1,572,864
#define BANK_ELEMS ((long long)NMEM * DIM) // 25,600,000

typedef float v2f __attribute__((ext_vector_type(2)));
typedef float v8f __attribute__((ext_vector_type(8)));

// ---------------------------------------------------------------------------
// Kernel 1: stream-copy one memory bank into the output buffer (float4).
// ---------------------------------------------------------------------------
__global__ void copy_bank(const float4* __restrict__ src, float4* __restrict__ dst,
                          long long n4) {
    long long stride = (long long)gridDim.x * blockDim.x;
    for (long long i = (long long)blockIdx.x * blockDim.x + threadIdx.x; i < n4; i += stride)
        dst[i] = src[i];
}

// ---------------------------------------------------------------------------
// Kernel 2: EMA + renormalize the B updated rows of one bank (1 wave per row).
// Reads ORIGINAL bank (input), writes into the already-copied output bank.
// ---------------------------------------------------------------------------
__global__ void ema_rows(const float* __restrict__ mem, const float* __restrict__ v,
                         const int* __restrict__ y, float* __restrict__ outmem) {
    const int b    = blockIdx.x;      // 0..BATCH-1
    const int lane = threadIdx.x;     // 0..31 (wave32)
    const int row  = y[b];
    const float4 m = ((const float4*)(mem + (size_t)row * DIM))[lane];
    const float4 x = ((const float4*)(v   + (size_t)b   * DIM))[lane];
    float4 p;
    p.x = m.x * 0.5f + x.x * 0.5f;
    p.y = m.y * 0.5f + x.y * 0.5f;
    p.z = m.z * 0.5f + x.z * 0.5f;
    p.w = m.w * 0.5f + x.w * 0.5f;
    float ss = p.x * p.x + p.y * p.y + p.z * p.z + p.w * p.w;
    #pragma unroll
    for (int off = 16; off >= 1; off >>= 1)
        ss += __shfl_xor(ss, off, 32);
    const float inv = 1.0f / sqrtf(ss);
    p.x *= inv; p.y *= inv; p.z *= inv; p.w *= inv;
    ((float4*)(outmem + (size_t)row * DIM))[lane] = p;
}

// ---------------------------------------------------------------------------
// Kernel 3: gathered GEMV via V_WMMA_F32_16X16X4_F32.
// Block = 128 threads (4 waves); blockIdx.y = b; each wave owns one 16-k tile.
// B operand (4x16 f32): columns 0..3 = {l, ab, ori, comp}[b], columns 4..15 = 0,
// staged in LDS and shared by all four WMMA accumulation chains.
// ---------------------------------------------------------------------------
__global__ void __launch_bounds__(128)
nce_dots_wmma(const float* __restrict__ lvec,  const float* __restrict__ abvec,
              const float* __restrict__ orivec, const float* __restrict__ compvec,
              const int*   __restrict__ idx,
              const float* __restrict__ ml,   const float* __restrict__ mab,
              const float* __restrict__ mori, const float* __restrict__ mcomp,
              float* __restrict__ outs) {
    __shared__ float svec[5 * DIM];   // rows: l, ab, ori, comp, zeros

    const int b   = blockIdx.y;
    const int tid = threadIdx.x;

    // Stage the four batch vectors + a zero row.
    svec[0 * DIM + tid] = lvec  [b * DIM + tid];
    svec[1 * DIM + tid] = abvec [b * DIM + tid];
    svec[2 * DIM + tid] = orivec[b * DIM + tid];
    svec[3 * DIM + tid] = compvec[b * DIM + tid];
    svec[4 * DIM + tid] = 0.0f;
    __syncthreads();

    const int wave  = tid >> 5;
    const int lane  = tid & 31;
    const int j     = lane & 15;   // matrix column / row-within-tile selector
    const int half  = lane >> 4;   // K-pair selector for A/B operands
    const int ktile = blockIdx.x * 4 + wave;
    const int k0    = ktile * 16;

    // One shared row index per (b,k): all four banks gather the same row.
    const int r = idx[b * K1 + k0 + j];
    const float* rl  = ml    + (size_t)r * DIM;
    const float* rab = mab   + (size_t)r * DIM;
    const float* ro  = mori  + (size_t)r * DIM;
    const float* rc  = mcomp + (size_t)r * DIM;

    const int vrow = (j < 4) ? j : 4;                 // columns >=4 read zeros
    const float* bvp = &svec[vrow * DIM + half * 2];  // per-lane B source
    const int aoff = half * 2;                        // per-lane A K-offset

    v8f accL = {}, accAB = {}, accORI = {}, accCOMP = {};

    #pragma unroll 8
    for (int kk = 0; kk < 32; ++kk) {
        const int o = kk * 4;
        const v2f bv = *(const v2f*)(bvp + o);            // ds_load_b64
        const v2f a0 = *(const v2f*)(rl  + o + aoff);     // global_load_b64 gathers
        const v2f a1 = *(const v2f*)(rab + o + aoff);
        const v2f a2 = *(const v2f*)(ro  + o + aoff);
        const v2f a3 = *(const v2f*)(rc  + o + aoff);
        accL    = __builtin_amdgcn_wmma_f32_16x16x4_f32(false, a0, false, bv, (short)0, accL,    false, false);
        accAB   = __builtin_amdgcn_wmma_f32_16x16x4_f32(false, a1, false, bv, (short)0, accAB,   false, false);
        accORI  = __builtin_amdgcn_wmma_f32_16x16x4_f32(false, a2, false, bv, (short)0, accORI,  false, false);
        accCOMP = __builtin_amdgcn_wmma_f32_16x16x4_f32(false, a3, false, bv, (short)0, accCOMP, false, false);
    }

    // D layout: VGPR v -> M = v + 8*half (row within tile), N = j (column).
    // outs stack order: [l_ori, ab_l, ori_ab, ab_comp, l_comp, comp_ori]
    const long long base = (long long)b * K1 + k0 + half * 8;
    const long long S = (long long)BATCH * K1;
    #pragma unroll
    for (int v = 0; v < 8; ++v) {
        if (j == 0) {
            outs[0 * S + base + v] = accORI[v]  / TNORM;  // dot(w_ori, l)
            outs[4 * S + base + v] = accCOMP[v] / TNORM;  // dot(w_comp, l)
        } else if (j == 1) {
            outs[1 * S + base + v] = accL[v]    / TNORM;  // dot(w_l, ab)
            outs[3 * S + base + v] = accCOMP[v] / TNORM;  // dot(w_comp, ab)
        } else if (j == 2) {
            outs[2 * S + base + v] = accAB[v]   / TNORM;  // dot(w_ab, ori)
        } else if (j == 3) {
            outs[5 * S + base + v] = accORI[v]  / TNORM;  // dot(w_ori, comp)
        }
    }
}

// ---------------------------------------------------------------------------
extern "C" void kernel_launch(void* const* d_in, const int* in_sizes, int n_in,
                              void* d_out, int out_size, void* d_ws, size_t ws_size,
                              hipStream_t stream) {
    const float* l     = (const float*)d_in[0];
    const float* ab    = (const float*)d_in[1];
    const float* ori   = (const float*)d_in[2];
    const float* comp  = (const float*)d_in[3];
    const int*   y     = (const int*)  d_in[4];
    const int*   idx   = (const int*)  d_in[5];
    const float* ml    = (const float*)d_in[6];
    const float* mab   = (const float*)d_in[7];
    const float* mori  = (const float*)d_in[8];
    const float* mcomp = (const float*)d_in[9];

    float* outs  = (float*)d_out;
    float* o_ml    = outs + OUTS_ELEMS;
    float* o_mab   = o_ml   + BANK_ELEMS;
    float* o_mori  = o_mab  + BANK_ELEMS;
    float* o_mcomp = o_mori + BANK_ELEMS;

    // 1) Stream-copy the four banks into the output (bandwidth-bound, ~820MB).
    const long long n4 = BANK_ELEMS / 4;
    copy_bank<<<2048, 256, 0, stream>>>((const float4*)ml,    (float4*)o_ml,    n4);
    copy_bank<<<2048, 256, 0, stream>>>((const float4*)mab,   (float4*)o_mab,   n4);
    copy_bank<<<2048, 256, 0, stream>>>((const float4*)mori,  (float4*)o_mori,  n4);
    copy_bank<<<2048, 256, 0, stream>>>((const float4*)mcomp, (float4*)o_mcomp, n4);

    // 2) EMA + renorm the B touched rows of each bank (reads original inputs).
    ema_rows<<<BATCH, 32, 0, stream>>>(ml,    l,    y, o_ml);
    ema_rows<<<BATCH, 32, 0, stream>>>(mab,   ab,   y, o_mab);
    ema_rows<<<BATCH, 32, 0, stream>>>(mori,  ori,  y, o_mori);
    ema_rows<<<BATCH, 32, 0, stream>>>(mcomp, comp, y, o_mcomp);

    // 3) Gathered GEMVs with f32 WMMA: grid = (K1/64 tiles-of-4, B).
    nce_dots_wmma<<<dim3(K1 / 64, BATCH), 128, 0, stream>>>(
        l, ab, ori, comp, idx, ml, mab, mori, mcomp, outs);
}